// VanillaRNN_39986145526476
// MI455X (gfx1250) — compile-verified
//
#include <hip/hip_runtime.h>
#include <math.h>

typedef __attribute__((ext_vector_type(2))) float v2f;
typedef __attribute__((ext_vector_type(4))) float v4f;
typedef __attribute__((ext_vector_type(8))) float v8f;

#define BATCH 4096
#define KDIM  2048
#define HID   2048
#define NCLS  10

#define BM 128
#define BN 128
#define BK 16
#define A_ST 20            // LDS A row stride (floats): 80B, 8B-aligned, conflict-free b64 reads
#define B_ST 132           // LDS B row stride (floats): 528B, 16B-aligned for b128 stores
#define A_TILE (BM * A_ST) // 2560 floats
#define B_TILE (BK * B_ST) // 2112 floats
#define NSTAGE 256         // 2 phases x (2048/16)

// h = tanh(x @ W_hx + h0 @ W_hh + b_h): fused two-GEMM fp32 WMMA, double-buffered LDS.
__global__ __launch_bounds__(256)
void rnn_h_kernel(const float* __restrict__ x,
                  const float* __restrict__ W_hx,
                  const float* __restrict__ W_hh,
                  const float* __restrict__ b_h,
                  const float* __restrict__ h0,
                  float* __restrict__ out_h)
{
    __shared__ float As[2 * A_TILE];
    __shared__ float Bs[2 * B_TILE];

    const int tid  = threadIdx.x;
    const int lane = tid & 31;          // wave32
    const int wid  = tid >> 5;          // 8 waves
    const int ln   = lane & 15;         // m (A frag) / n (B,C frag) index
    const int kb   = (lane >> 4) << 1;  // 0 or 2 (ISA fp32 16x4 A/B layout)

    const int wm = (wid & 1) * 64;      // wave M offset: 2 waves down
    const int wn = (wid >> 1) * 32;     // wave N offset: 4 waves across

    const int block_m = blockIdx.x * BM;
    const int block_n = blockIdx.y * BN;

    // per-thread staging coordinates
    const int a_row = tid >> 2;               // 0..63 (and +64)
    const int a_c   = (tid & 3) * 4;          // float4 column within BK
    const int b_row = tid >> 5;               // 0..7 (and +8)
    const int b_c   = (tid & 31) * 4;         // float4 column within BN

    v8f acc[4][2];
    #pragma unroll
    for (int mi = 0; mi < 4; ++mi)
        #pragma unroll
        for (int nj = 0; nj < 2; ++nj)
            #pragma unroll
            for (int r = 0; r < 8; ++r)
                acc[mi][nj][r] = 0.0f;

    v4f pa0, pa1, pb0, pb1;   // in-flight stage registers

    auto load_stage = [&](int s) {
        const int phase = s >> 7;             // 0: x@W_hx, 1: h0@W_hh
        const int k0    = (s & 127) * BK;
        const float* __restrict__ Ag = phase ? h0   : x;
        const float* __restrict__ Bg = phase ? W_hh : W_hx;
        pa0 = *(const v4f*)(Ag + (size_t)(block_m + a_row)      * KDIM + k0 + a_c);
        pa1 = *(const v4f*)(Ag + (size_t)(block_m + a_row + 64) * KDIM + k0 + a_c);
        pb0 = *(const v4f*)(Bg + (size_t)(k0 + b_row)     * HID + block_n + b_c);
        pb1 = *(const v4f*)(Bg + (size_t)(k0 + b_row + 8) * HID + block_n + b_c);
    };

    auto store_stage = [&](int buf) {
        float* __restrict__ a = As + buf * A_TILE;
        float* __restrict__ b = Bs + buf * B_TILE;
        v2f t;
        float* sp0 = a + a_row * A_ST + a_c;
        t.x = pa0.x; t.y = pa0.y; *(v2f*)(sp0)     = t;
        t.x = pa0.z; t.y = pa0.w; *(v2f*)(sp0 + 2) = t;
        float* sp1 = a + (a_row + 64) * A_ST + a_c;
        t.x = pa1.x; t.y = pa1.y; *(v2f*)(sp1)     = t;
        t.x = pa1.z; t.y = pa1.w; *(v2f*)(sp1 + 2) = t;
        *(v4f*)(b + b_row * B_ST + b_c)       = pb0;
        *(v4f*)(b + (b_row + 8) * B_ST + b_c) = pb1;
    };

    auto compute_stage = [&](int buf) {
        const float* __restrict__ a = As + buf * A_TILE;
        const float* __restrict__ b = Bs + buf * B_TILE;
        #pragma unroll
        for (int kk = 0; kk < BK; kk += 4) {
            v2f af[4];
            #pragma unroll
            for (int mi = 0; mi < 4; ++mi)
                af[mi] = *(const v2f*)&a[(wm + mi * 16 + ln) * A_ST + kk + kb];
            v2f bf[2];
            #pragma unroll
            for (int nj = 0; nj < 2; ++nj) {
                bf[nj].x = b[(kk + kb)     * B_ST + wn + nj * 16 + ln];
                bf[nj].y = b[(kk + kb + 1) * B_ST + wn + nj * 16 + ln];
            }
            #pragma unroll
            for (int mi = 0; mi < 4; ++mi)
                #pragma unroll
                for (int nj = 0; nj < 2; ++nj)
                    acc[mi][nj] = __builtin_amdgcn_wmma_f32_16x16x4_f32(
                        false, af[mi], false, bf[nj],
                        (short)0, acc[mi][nj], false, false);
        }
    };

    // ---- software pipeline: 1 barrier per stage, loads overlap compute ----
    load_stage(0);
    store_stage(0);
    __syncthreads();

    #pragma unroll 1
    for (int s = 0; s < NSTAGE - 1; ++s) {
        load_stage(s + 1);          // global loads for next stage (in regs)
        compute_stage(s & 1);       // 32 WMMAs hide the load latency
        store_stage((s + 1) & 1);   // other buffer: no race with readers of buf s&1
        __syncthreads();
    }
    compute_stage((NSTAGE - 1) & 1);

    // ---- epilogue: + b_h, tanh, store h ----
    const int half8 = (lane >> 4) * 8;   // C/D row = r + 8*(lane>=16)
    #pragma unroll
    for (int mi = 0; mi < 4; ++mi) {
        #pragma unroll
        for (int nj = 0; nj < 2; ++nj) {
            const int gcol = block_n + wn + nj * 16 + ln;
            #pragma unroll
            for (int r = 0; r < 8; ++r) {
                const int grow = block_m + wm + mi * 16 + r + half8;
                const size_t idx = (size_t)grow * HID + gcol;
                out_h[idx] = tanhf(acc[mi][nj][r] + b_h[idx]);
            }
        }
    }
}

// p = h @ W_ph  (N=10): one block per row, strided partials + wave32 shuffle reduce.
__global__ __launch_bounds__(256)
void rnn_p_kernel(const float* __restrict__ h,
                  const float* __restrict__ W_ph,
                  float* __restrict__ out_p)
{
    __shared__ float sm[8][NCLS];
    const int row  = blockIdx.x;
    const int tid  = threadIdx.x;
    const int lane = tid & 31;
    const int wid  = tid >> 5;

    float acc[NCLS];
    #pragma unroll
    for (int c = 0; c < NCLS; ++c) acc[c] = 0.0f;

    const float* __restrict__ hp = h + (size_t)row * HID;
    for (int k = tid; k < HID; k += 256) {
        const float hv = hp[k];
        const float* wp = W_ph + (size_t)k * NCLS;
        #pragma unroll
        for (int c = 0; c < NCLS; ++c) acc[c] += hv * wp[c];
    }
    #pragma unroll
    for (int c = 0; c < NCLS; ++c) {
        #pragma unroll
        for (int off = 16; off > 0; off >>= 1)
            acc[c] += __shfl_down(acc[c], off, 32);
    }
    if (lane == 0) {
        #pragma unroll
        for (int c = 0; c < NCLS; ++c) sm[wid][c] = acc[c];
    }
    __syncthreads();
    if (tid < NCLS) {
        float s = 0.0f;
        #pragma unroll
        for (int w = 0; w < 8; ++w) s += sm[w][tid];
        out_p[(size_t)row * NCLS + tid] = s;
    }
}

extern "C" void kernel_launch(void* const* d_in, const int* in_sizes, int n_in,
                              void* d_out, int out_size, void* d_ws, size_t ws_size,
                              hipStream_t stream) {
    const float* x    = (const float*)d_in[0];
    const float* W_hx = (const float*)d_in[1];
    const float* W_hh = (const float*)d_in[2];
    const float* W_ph = (const float*)d_in[3];
    const float* b_h  = (const float*)d_in[4];
    const float* h0   = (const float*)d_in[5];

    float* out_p = (float*)d_out;                     // [4096, 10] first
    float* out_h = out_p + (size_t)BATCH * NCLS;      // then [4096, 2048]

    dim3 grid(BATCH / BM, HID / BN);                  // 32 x 16 blocks
    rnn_h_kernel<<<grid, 256, 0, stream>>>(x, W_hx, W_hh, b_h, h0, out_h);
    rnn_p_kernel<<<BATCH, 256, 0, stream>>>(out_h, W_ph, out_p);
}